// GATlayer_81595788689719
// MI455X (gfx1250) — compile-verified
//
#include <hip/hip_runtime.h>

#define N_IN 128          // input channels
#define HC   128          // H*C = 2*64 output channels
#define NEG_SLOPE 0.2f

typedef __attribute__((ext_vector_type(2))) float v2f;
typedef __attribute__((ext_vector_type(8))) float v8f;
typedef __attribute__((ext_vector_type(4))) unsigned int u32x4;
typedef __attribute__((ext_vector_type(8))) int i32x8;
typedef __attribute__((ext_vector_type(4))) int i32x4;

// ---------------------------------------------------------------------------
// Kernel 1: x = h_node @ W via V_WMMA_F32_16X16X4_F32.
// Block = 128 threads (4 waves). Each block computes a 64x128 output tile:
// wave w handles rows [blk*64 + 16w, +16), all 8 N-tiles of 16.
// W (128x128 fp32 = 64KB) is staged into LDS by the Tensor Data Mover
// (one tensor_load_to_lds from wave 0), falling back to a cooperative copy.
// ---------------------------------------------------------------------------
__global__ __launch_bounds__(128) void gat_gemm_wmma(
    const float* __restrict__ A,   // [Nn,128] h_node
    const float* __restrict__ W,   // [128,128]
    float* __restrict__ X,         // [Nn,128] out
    int Nn)
{
    __shared__ float sW[N_IN * HC];   // 64 KB, row-major [k][n]; LDS offset 0

#if defined(__gfx1250__) && __has_builtin(__builtin_amdgcn_tensor_load_to_lds) && \
    __has_builtin(__builtin_amdgcn_s_wait_tensorcnt)
    // --- TDM staging: one DMA descriptor moves the whole 128x128 f32 tile ---
    if (threadIdx.x < 32) {            // wave 0 issues (TDM ignores EXEC)
        const unsigned long long ga = (unsigned long long)(uintptr_t)W;
        u32x4 g0;
        g0.x = 1u;                                     // count=1, user descriptor
        g0.y = 0u;                                     // lds_addr = 0 (sW base)
        g0.z = (unsigned int)(ga & 0xFFFFFFFFu);       // global_addr[31:0]
        g0.w = (unsigned int)((ga >> 32) & 0x01FFFFFFu)// global_addr[56:32]
             | (2u << 30);                             // type = 2 ("image")
        i32x8 g1;
        g1[0] = 2 << 16;          // data_size = 4 bytes; no multicast/pad/iterate
        g1[1] = 128 << 16;        // tensor_dim0 = 128 (lo16 in [31:16])
        g1[2] = 128 << 16;        // tensor_dim0 hi = 0 | tensor_dim1 = 128
        g1[3] = 128 << 16;        // tensor_dim1 hi = 0 | tile_dim0 = 128
        g1[4] = 128;              // tile_dim1 = 128, tile_dim2 = 0 (2D)
        g1[5] = 128;              // tensor_dim0_stride = 128 elements
        g1[6] = 0;                // stride hi / dim1_stride lo (unused for 2D)
        g1[7] = 0;
        const i32x4 z4 = {0, 0, 0, 0};
#if defined(__clang_major__) && (__clang_major__ >= 23)
        const i32x8 z8 = {0, 0, 0, 0, 0, 0, 0, 0};
        __builtin_amdgcn_tensor_load_to_lds(g0, g1, z4, z4, z8, 0);
#else
        __builtin_amdgcn_tensor_load_to_lds(g0, g1, z4, z4, 0);
#endif
        __builtin_amdgcn_s_wait_tensorcnt(0);
    }
    __syncthreads();
    // The TDM descriptor only carries an integer LDS offset, so the compiler
    // cannot see that sW was written. Escape the pointer with a memory clobber
    // so the subsequent ds_loads are not folded to undef (round-2 bug: all B
    // fragments CSE'd into one junk register pair).
    {
        float* p = sW;
        asm volatile("" : "+v"(p) : : "memory");
    }
#else
    for (int i = threadIdx.x; i < (N_IN * HC) / 4; i += 128) {
        ((float4*)sW)[i] = ((const float4*)W)[i];
    }
    __syncthreads();
#endif

    const int wave  = threadIdx.x >> 5;
    const int lane  = threadIdx.x & 31;
    const int mrow  = lane & 15;            // M index within tile (A), N index (B/C)
    const int khalf = (lane >> 4) << 1;     // 0 for lanes 0-15, 2 for lanes 16-31
    const int rowBase = blockIdx.x * 64 + wave * 16;

    v8f acc[8];
#pragma unroll
    for (int t = 0; t < 8; ++t) acc[t] = (v8f){0.f,0.f,0.f,0.f,0.f,0.f,0.f,0.f};

    // clamp A row so the wave never diverges before WMMA (EXEC must be all-1s);
    // out-of-range rows are simply not stored.
    const int arow = min(rowBase + mrow, Nn - 1);
    const float* __restrict__ Ar = A + (size_t)arow * N_IN;

    for (int k0 = 0; k0 < N_IN; k0 += 4) {
        // A fragment (16x4 f32): lane<16 -> K=k0,k0+1 ; lane>=16 -> K=k0+2,k0+3
        v2f a;
        a.x = Ar[k0 + khalf];
        a.y = Ar[k0 + khalf + 1];
#pragma unroll
        for (int t = 0; t < 8; ++t) {
            const int n = t * 16 + mrow;
            v2f b;                           // B fragment (4x16 f32), mirrored layout
            b.x = sW[(k0 + khalf)     * HC + n];
            b.y = sW[(k0 + khalf + 1) * HC + n];
            acc[t] = __builtin_amdgcn_wmma_f32_16x16x4_f32(
                false, a, false, b, (short)0, acc[t], false, false);
        }
    }

    // C/D layout: VGPR r -> M = r (lanes 0-15) / r+8 (lanes 16-31), N = lane&15
    const int rofs = (lane >> 4) << 3;
    if (rowBase + 15 < Nn) {
        // fast path: whole 16-row tile in range, no per-store exec juggling
#pragma unroll
        for (int t = 0; t < 8; ++t) {
            const int col = t * 16 + mrow;
            float* __restrict__ Xp = X + (size_t)(rowBase + rofs) * HC + col;
#pragma unroll
            for (int r = 0; r < 8; ++r) Xp[(size_t)r * HC] = acc[t][r];
        }
    } else {
        // tail block only (1 of ~782): guarded stores
#pragma unroll
        for (int t = 0; t < 8; ++t) {
            const int col = t * 16 + mrow;
#pragma unroll
            for (int r = 0; r < 8; ++r) {
                const int row = rowBase + r + rofs;
                if (row < Nn) X[(size_t)row * HC + col] = acc[t][r];
            }
        }
    }
}

// ---------------------------------------------------------------------------
// Kernel 2: per-node attention terms. One wave per node; lane l covers
// channels 4l..4l+3 (head = l>>4). Shuffle-reduce within each 16-lane half.
// ---------------------------------------------------------------------------
__global__ __launch_bounds__(256) void gat_attn_terms(
    const float* __restrict__ X,
    const float* __restrict__ att_src,   // [128] flat (H*C)
    const float* __restrict__ att_dst,   // [128]
    float* __restrict__ a_s,             // [Nn*2]
    float* __restrict__ a_d,             // [Nn*2]
    int Nn)
{
    const int gw   = (blockIdx.x * blockDim.x + threadIdx.x) >> 5;
    const int lane = threadIdx.x & 31;
    if (gw >= Nn) return;

    const float4 xv = ((const float4*)(X + (size_t)gw * HC))[lane];
    const float4 as = ((const float4*)att_src)[lane];
    const float4 ad = ((const float4*)att_dst)[lane];

    float ps = xv.x*as.x + xv.y*as.y + xv.z*as.z + xv.w*as.w;
    float pd = xv.x*ad.x + xv.y*ad.y + xv.z*ad.z + xv.w*ad.w;

#pragma unroll
    for (int off = 8; off >= 1; off >>= 1) {   // reduce within 16-lane groups
        ps += __shfl_xor(ps, off, 32);
        pd += __shfl_xor(pd, off, 32);
    }
    if (lane == 0)  { a_s[gw*2 + 0] = ps; a_d[gw*2 + 0] = pd; }
    if (lane == 16) { a_s[gw*2 + 1] = ps; a_d[gw*2 + 1] = pd; }
}

// ---------------------------------------------------------------------------
// Kernel 3: init accumulators (required every launch; harness poisons buffers)
// ---------------------------------------------------------------------------
__global__ __launch_bounds__(256) void gat_init(
    float* __restrict__ out, float* __restrict__ m, float* __restrict__ den, int Nn)
{
    const int i = blockIdx.x * blockDim.x + threadIdx.x;
    if (i < Nn * HC) out[i] = 0.f;
    if (i < Nn * 2) { m[i] = -__builtin_inff(); den[i] = 0.f; }
}

// ---------------------------------------------------------------------------
// Edge helpers
// ---------------------------------------------------------------------------
__device__ __forceinline__ void edge_sd(const int* __restrict__ ei, int E, int idx,
                                        int& s, int& d)
{
    if (idx < E) { s = ei[idx]; d = ei[E + idx]; }
    else         { s = d = idx - E; }               // appended self-loops
}

__device__ __forceinline__ float leaky(float e)
{
    return e > 0.f ? e : NEG_SLOPE * e;
}

// exact IEEE float atomic-max via signed/unsigned integer atomics
__device__ __forceinline__ void atomicMaxF(float* addr, float v)
{
    if (v >= 0.f) atomicMax((int*)addr, __float_as_int(v));
    else          atomicMin((unsigned int*)addr, __float_as_uint(v));
}

// ---------------------------------------------------------------------------
// Kernel 4: segment max of leaky-relu scores over destination nodes
// ---------------------------------------------------------------------------
__global__ __launch_bounds__(256) void gat_edge_max(
    const int* __restrict__ ei, int E, int TE,
    const float* __restrict__ a_s, const float* __restrict__ a_d,
    float* __restrict__ m)
{
    const int i = blockIdx.x * blockDim.x + threadIdx.x;
    if (i >= TE) return;
    int s, d; edge_sd(ei, E, i, s, d);
#pragma unroll
    for (int h = 0; h < 2; ++h) {
        const float e = leaky(a_s[s*2 + h] + a_d[d*2 + h]);
        atomicMaxF(&m[d*2 + h], e);
    }
}

// ---------------------------------------------------------------------------
// Kernel 5: softmax denominator
// ---------------------------------------------------------------------------
__global__ __launch_bounds__(256) void gat_edge_den(
    const int* __restrict__ ei, int E, int TE,
    const float* __restrict__ a_s, const float* __restrict__ a_d,
    const float* __restrict__ m, float* __restrict__ den)
{
    const int i = blockIdx.x * blockDim.x + threadIdx.x;
    if (i >= TE) return;
    int s, d; edge_sd(ei, E, i, s, d);
#pragma unroll
    for (int h = 0; h < 2; ++h) {
        const float e = leaky(a_s[s*2 + h] + a_d[d*2 + h]);
        atomicAdd(&den[d*2 + h], __expf(e - m[d*2 + h]));
    }
}

// ---------------------------------------------------------------------------
// Kernel 6: weighted scatter. One wave per edge; lane l owns channels
// 4l..4l+3 (head = l>>4). Coalesced float4 gather of x[src], 4 fp32
// global atomic adds into out[dst].
// ---------------------------------------------------------------------------
__global__ __launch_bounds__(256) void gat_edge_scatter(
    const int* __restrict__ ei, int E, int TE,
    const float* __restrict__ a_s, const float* __restrict__ a_d,
    const float* __restrict__ m, const float* __restrict__ den,
    const float* __restrict__ X, float* __restrict__ out)
{
    const int e    = (blockIdx.x * blockDim.x + threadIdx.x) >> 5;
    const int lane = threadIdx.x & 31;
    if (e >= TE) return;
    int s, d; edge_sd(ei, E, e, s, d);

    const int h = lane >> 4;
    const float sc    = leaky(a_s[s*2 + h] + a_d[d*2 + h]);
    const float alpha = __expf(sc - m[d*2 + h]) / den[d*2 + h];

    const float4 xv = ((const float4*)(X + (size_t)s * HC))[lane];
    float* op = out + (size_t)d * HC + lane * 4;
    atomicAdd(op + 0, alpha * xv.x);
    atomicAdd(op + 1, alpha * xv.y);
    atomicAdd(op + 2, alpha * xv.z);
    atomicAdd(op + 3, alpha * xv.w);
}

// ---------------------------------------------------------------------------
// Kernel 7: out = elu(acc + bias)
// ---------------------------------------------------------------------------
__global__ __launch_bounds__(256) void gat_finalize(
    float* __restrict__ out, const float* __restrict__ bias, int total)
{
    const int i = blockIdx.x * blockDim.x + threadIdx.x;
    if (i >= total) return;
    const float v = out[i] + bias[i & (HC - 1)];
    out[i] = v > 0.f ? v : (__expf(v) - 1.f);
}

// ---------------------------------------------------------------------------
// Host-side launcher
// ---------------------------------------------------------------------------
extern "C" void kernel_launch(void* const* d_in, const int* in_sizes, int n_in,
                              void* d_out, int out_size, void* d_ws, size_t ws_size,
                              hipStream_t stream)
{
    const float* h_node  = (const float*)d_in[0];
    const int*   ei      = (const int*)  d_in[1];
    const float* W       = (const float*)d_in[2];
    const float* att_src = (const float*)d_in[3];
    const float* att_dst = (const float*)d_in[4];
    const float* bias    = (const float*)d_in[5];
    float* out = (float*)d_out;

    const int Nn = in_sizes[0] / N_IN;     // nodes
    const int E  = in_sizes[1] / 2;        // edges before self-loops
    const int TE = E + Nn;                 // with self-loops

    // workspace layout (floats)
    float* X   = (float*)d_ws;                       // [Nn*128]
    float* a_s = X   + (size_t)Nn * HC;              // [Nn*2]
    float* a_d = a_s + (size_t)Nn * 2;               // [Nn*2]
    float* m   = a_d + (size_t)Nn * 2;               // [Nn*2]
    float* den = m   + (size_t)Nn * 2;               // [Nn*2]

    // 1: projection GEMM (WMMA + TDM-staged W)
    gat_gemm_wmma<<<(Nn + 63) / 64, 128, 0, stream>>>(h_node, W, X, Nn);

    // 2: per-node attention terms (one wave per node)
    gat_attn_terms<<<(Nn * 32 + 255) / 256, 256, 0, stream>>>(X, att_src, att_dst,
                                                              a_s, a_d, Nn);

    // 3: init out / m / den
    gat_init<<<(Nn * HC + 255) / 256, 256, 0, stream>>>(out, m, den, Nn);

    // 4: segment max
    gat_edge_max<<<(TE + 255) / 256, 256, 0, stream>>>(ei, E, TE, a_s, a_d, m);

    // 5: softmax denominator
    gat_edge_den<<<(TE + 255) / 256, 256, 0, stream>>>(ei, E, TE, a_s, a_d, m, den);

    // 6: weighted scatter (one wave per edge)
    {
        const long long threads = (long long)TE * 32;
        gat_edge_scatter<<<(unsigned)((threads + 255) / 256), 256, 0, stream>>>(
            ei, E, TE, a_s, a_d, m, den, X, out);
    }

    // 7: bias + ELU
    gat_finalize<<<(Nn * HC + 255) / 256, 256, 0, stream>>>(out, bias, Nn * HC);
}